// SingleGNNLayer_4260607557816
// MI455X (gfx1250) — compile-verified
//
#include <hip/hip_runtime.h>
#include <math.h>

typedef __attribute__((ext_vector_type(2))) float v2f;
typedef __attribute__((ext_vector_type(8))) float v8f;

#define N_NODES   50000
#define N_EDGES   800000
#define IN_DIM    256
#define OUT_DIM   256
#define EDGE_DIM  128
#define NEG_SLOPE 0.01f

// ---------------- init: zero out, m=-inf, denom=0 ----------------
__global__ void k_init(float* __restrict__ out, float* __restrict__ m,
                       float* __restrict__ denom) {
  int i = blockIdx.x * blockDim.x + threadIdx.x;
  if (i < N_NODES * OUT_DIM) out[i] = 0.0f;
  if (i < N_NODES) { m[i] = -INFINITY; denom[i] = 0.0f; }
}

// ---------------- h = x @ Wn^T + bn via fp32 WMMA 16x16x4 ----------------
// One wave computes one 16x16 output tile. Block = 8 waves (8 consecutive
// m-tiles, same n-tile). B panel (Wn columns n0..n0+15, K=256) staged in LDS.
__global__ __launch_bounds__(256) void k_node_gemm(const float* __restrict__ x,
                                                   const float* __restrict__ Wn,
                                                   const float* __restrict__ bn,
                                                   float* __restrict__ h) {
  __shared__ float lb[IN_DIM * 16];            // lb[k*16 + n] = Wn[n0+n, k]
  const int n0 = blockIdx.y * 16;
  for (int idx = threadIdx.x; idx < IN_DIM * 16; idx += 256) {
    int n = idx >> 8;                          // 0..15
    int k = idx & 255;                         // 0..255
    lb[k * 16 + n] = Wn[(n0 + n) * IN_DIM + k];
  }
  __syncthreads();

  const int wave   = threadIdx.x >> 5;
  const int lane   = threadIdx.x & 31;
  const int half   = lane >> 4;                // 0: K pair {0,1}; 1: {2,3}
  const int lanelo = lane & 15;
  const int mtile  = blockIdx.x * 8 + wave;
  if (mtile >= N_NODES / 16) return;           // wave-uniform exit, EXEC stays full
  const int m0 = mtile * 16;

  const float* arow = x + (size_t)(m0 + lanelo) * IN_DIM + 2 * half;
  v8f acc = {};
  for (int k = 0; k < IN_DIM; k += 4) {
    v2f a = *(const v2f*)(arow + k);           // A[m0+lanelo, k+2h .. +1]
    v2f b;
    b.x = lb[(k + 2 * half)     * 16 + lanelo];// B[k+2h,   n0+lanelo]
    b.y = lb[(k + 2 * half + 1) * 16 + lanelo];// B[k+2h+1, n0+lanelo]
    acc = __builtin_amdgcn_wmma_f32_16x16x4_f32(false, a, false, b,
                                                (short)0, acc, false, false);
  }
  const float bias = bn[n0 + lanelo];
  float* ocol = h + n0 + lanelo;
#pragma unroll
  for (int r = 0; r < 8; ++r)
    ocol[(size_t)(m0 + 8 * half + r) * OUT_DIM] = acc[r] + bias;
}

// ---------------- s[n] = dot(h[n,:], attn) ----------------
__global__ __launch_bounds__(256) void k_node_dot(const float* __restrict__ h,
                                                  const float* __restrict__ attn,
                                                  float* __restrict__ s) {
  int node = (int)((blockIdx.x * blockDim.x + threadIdx.x) >> 5);
  int lane = threadIdx.x & 31;
  if (node >= N_NODES) return;
  const float* row = h + (size_t)node * OUT_DIM;
  float acc = 0.0f;
#pragma unroll
  for (int j = 0; j < 8; ++j) { int c = lane + 32 * j; acc += row[c] * attn[c]; }
  for (int off = 16; off; off >>= 1) acc += __shfl_xor(acc, off, 32);
  if (lane == 0) s[node] = acc;
}

// ---------------- w[k] = sum_c We[c,k]*attn[c];  wconst = be . attn ----------
__global__ void k_edge_vec(const float* __restrict__ We, const float* __restrict__ be,
                           const float* __restrict__ attn,
                           float* __restrict__ w, float* __restrict__ wconst) {
  int k = threadIdx.x;                         // 128 threads
  float acc = 0.0f;
  for (int c = 0; c < OUT_DIM; ++c) acc += We[c * EDGE_DIM + k] * attn[c];
  w[k] = acc;
  if (k == 0) {
    float cb = 0.0f;
    for (int c = 0; c < OUT_DIM; ++c) cb += be[c] * attn[c];
    *wconst = cb;
  }
}

// ---------------- per-edge logit + segment max ----------------
__global__ __launch_bounds__(256) void k_edge_logit(const float* __restrict__ edge_attr,
                                                    const int* __restrict__ ei,
                                                    const float* __restrict__ s,
                                                    const float* __restrict__ w,
                                                    const float* __restrict__ wconst,
                                                    float* __restrict__ logit,
                                                    float* __restrict__ m) {
  int e    = (int)((blockIdx.x * (long long)blockDim.x + threadIdx.x) >> 5);
  int lane = threadIdx.x & 31;
  if (e >= N_EDGES) return;
  const float4 ea = *(const float4*)(edge_attr + (size_t)e * EDGE_DIM + lane * 4);
  const float4 wv = *(const float4*)(w + lane * 4);
  float acc = ea.x * wv.x + ea.y * wv.y + ea.z * wv.z + ea.w * wv.w;
  for (int off = 16; off; off >>= 1) acc += __shfl_xor(acc, off, 32);
  if (lane == 0) {
    int src = ei[e], dst = ei[N_EDGES + e];
    float a = acc + s[src] + *wconst;
    a = a > 0.0f ? a : NEG_SLOPE * a;          // leaky relu
    logit[e] = a;
    __hip_atomic_fetch_max(&m[dst], a, __ATOMIC_RELAXED, __HIP_MEMORY_SCOPE_AGENT);
  }
}

// ---------------- exp(a - m[dst]) + segment sum ----------------
__global__ void k_edge_exp(const float* __restrict__ logit, const int* __restrict__ ei,
                           const float* __restrict__ m, float* __restrict__ expv,
                           float* __restrict__ denom) {
  int e = blockIdx.x * blockDim.x + threadIdx.x;
  if (e >= N_EDGES) return;
  int dst = ei[N_EDGES + e];
  float p = __expf(logit[e] - m[dst]);
  expv[e] = p;
  atomicAdd(&denom[dst], p);
}

// ---------------- weighted gather/scatter: out[dst] += coef * h[src] --------
__global__ __launch_bounds__(256) void k_scatter(const float* __restrict__ h,
                                                 const int* __restrict__ ei,
                                                 const float* __restrict__ expv,
                                                 const float* __restrict__ denom,
                                                 float* __restrict__ out) {
  int e    = (int)((blockIdx.x * (long long)blockDim.x + threadIdx.x) >> 5);
  int lane = threadIdx.x & 31;
  if (e >= N_EDGES) return;
  int src = ei[e], dst = ei[N_EDGES + e];
  float coef = expv[e] / (denom[dst] + 1e-16f);
  const float* hrow = h + (size_t)src * OUT_DIM;
  float* orow = out + (size_t)dst * OUT_DIM;
#pragma unroll
  for (int j = 0; j < 8; ++j) {
    int c = lane + 32 * j;                     // coalesced 128B atomic bursts
    atomicAdd(&orow[c], hrow[c] * coef);
  }
}

// ---------------- final relu ----------------
__global__ void k_relu(float* __restrict__ out) {
  int i = blockIdx.x * blockDim.x + threadIdx.x;
  if (i < N_NODES * OUT_DIM) out[i] = fmaxf(out[i], 0.0f);
}

extern "C" void kernel_launch(void* const* d_in, const int* in_sizes, int n_in,
                              void* d_out, int out_size, void* d_ws, size_t ws_size,
                              hipStream_t stream) {
  const float* x         = (const float*)d_in[0];
  const int*   ei        = (const int*)  d_in[1];   // [2, E]: row0=src, row1=dst
  const float* edge_attr = (const float*)d_in[2];
  const float* Wn        = (const float*)d_in[3];
  const float* bn        = (const float*)d_in[4];
  const float* We        = (const float*)d_in[5];
  const float* be        = (const float*)d_in[6];
  const float* attn      = (const float*)d_in[7];
  float* out = (float*)d_out;

  float* p      = (float*)d_ws;
  float* h      = p; p += (size_t)N_NODES * OUT_DIM;  // 12.8M
  float* s      = p; p += N_NODES;
  float* m      = p; p += N_NODES;
  float* denom  = p; p += N_NODES;
  float* w      = p; p += EDGE_DIM;
  float* wconst = p; p += 1;
  float* logit  = p; p += N_EDGES;
  float* expv   = p; p += N_EDGES;

  k_init<<<(N_NODES * OUT_DIM + 255) / 256, 256, 0, stream>>>(out, m, denom);

  dim3 gg(((N_NODES / 16) + 7) / 8, OUT_DIM / 16);    // 391 x 16 blocks
  k_node_gemm<<<gg, 256, 0, stream>>>(x, Wn, bn, h);

  k_node_dot<<<(N_NODES * 32 + 255) / 256, 256, 0, stream>>>(h, attn, s);
  k_edge_vec<<<1, 128, 0, stream>>>(We, be, attn, w, wconst);
  k_edge_logit<<<(N_EDGES * 32 + 255) / 256, 256, 0, stream>>>(edge_attr, ei, s, w,
                                                               wconst, logit, m);
  k_edge_exp<<<(N_EDGES + 255) / 256, 256, 0, stream>>>(logit, ei, m, expv, denom);
  k_scatter<<<(N_EDGES * 32 + 255) / 256, 256, 0, stream>>>(h, ei, expv, denom, out);
  k_relu<<<(N_NODES * OUT_DIM + 255) / 256, 256, 0, stream>>>(out);
}